// Model_59828894434002
// MI455X (gfx1250) — compile-verified
//
#include <hip/hip_runtime.h>
#include <math.h>

typedef float v2f __attribute__((ext_vector_type(2)));
typedef float v8f __attribute__((ext_vector_type(8)));

#define DD 16
#define RELS 20
#define NGRAPHS 128
#define ATTH 64
#define ATTO 32
#define WSTR 260   // padded LDS stride per relation: 260 % 64 = 4 banks apart, 16B-aligned rows

// ---------------- utility ----------------
__global__ void fill_zero(float* __restrict__ p, long n) {
    long i = (long)blockIdx.x * blockDim.x + threadIdx.x;
    long s = (long)gridDim.x * blockDim.x;
    for (; i < n; i += s) p[i] = 0.0f;
}

// ---------------- in-degree ----------------
__global__ void deg_kernel(const int* __restrict__ dst, float* __restrict__ deg, int E) {
    int e = blockIdx.x * blockDim.x + threadIdx.x;
    if (e < E) atomicAdd(&deg[dst[e]], 1.0f);
}

// ---------------- layer 0 edge scatter: agg[dst] += deg[src] * W0[etype] ----------------
__global__ void l0_edge_kernel(const float* __restrict__ deg,
                               const int* __restrict__ src,
                               const int* __restrict__ dst,
                               const int* __restrict__ et,
                               const float* __restrict__ W0,   // [R,1,16]
                               float* __restrict__ agg, int E) {
    __shared__ float Wl[RELS * 17];   // stride 17 -> conflict-free bank spread
    for (int i = threadIdx.x; i < RELS * DD; i += blockDim.x)
        Wl[(i >> 4) * 17 + (i & 15)] = W0[i];
    __syncthreads();
    int e = blockIdx.x * blockDim.x + threadIdx.x;
    if (e >= E) return;
    float dv = deg[src[e]];
    float* out = agg + (long)dst[e] * DD;
    const float* w = &Wl[et[e] * 17];
    #pragma unroll
    for (int j = 0; j < DD; ++j) atomicAdd(&out[j], dv * w[j]);
}

// ---------------- layer 0 node update (rank-1 self loop, scalar) ----------------
__global__ void l0_node_kernel(const float* __restrict__ agg,
                               const float* __restrict__ deg,
                               const float* __restrict__ loop0,  // [1,16]
                               const float* __restrict__ b0,
                               float* __restrict__ h, int N) {
    int i = blockIdx.x * blockDim.x + threadIdx.x;
    if (i >= N * DD) return;
    int n = i >> 4, j = i & 15;
    float v = agg[i] + deg[n] * loop0[j] + b0[j];
    h[i] = fmaxf(v, 0.0f);
}

// ---------------- layers 1/2 edge scatter: agg[dst] += h[src] @ W[etype] ----------------
__global__ void rgcn_edge_kernel(const float* __restrict__ h,
                                 const int* __restrict__ src,
                                 const int* __restrict__ dst,
                                 const int* __restrict__ et,
                                 const float* __restrict__ W,   // [R,16,16]
                                 float* __restrict__ agg, int E) {
    __shared__ float Wl[RELS * WSTR];
    for (int i = threadIdx.x; i < RELS * DD * DD; i += blockDim.x) {
        int r = i >> 8, o = i & 255;
        Wl[r * WSTR + o] = W[i];
    }
    __syncthreads();
    int e = blockIdx.x * blockDim.x + threadIdx.x;
    if (e >= E) return;
    int s = src[e], d = dst[e], r = et[e];
    const float4* hs = (const float4*)(h + (long)s * DD);  // L2-resident gather
    float4 hv4[4];
    hv4[0] = hs[0]; hv4[1] = hs[1]; hv4[2] = hs[2]; hv4[3] = hs[3];
    const float* hv = (const float*)hv4;
    const float4* wr = (const float4*)(&Wl[r * WSTR]);     // 16B-aligned ds_load_b128
    float4 m0 = {0,0,0,0}, m1 = m0, m2 = m0, m3 = m0;
    #pragma unroll
    for (int i = 0; i < DD; ++i) {
        float a = hv[i];
        float4 w0 = wr[i*4+0], w1 = wr[i*4+1], w2 = wr[i*4+2], w3 = wr[i*4+3];
        m0.x += a*w0.x; m0.y += a*w0.y; m0.z += a*w0.z; m0.w += a*w0.w;
        m1.x += a*w1.x; m1.y += a*w1.y; m1.z += a*w1.z; m1.w += a*w1.w;
        m2.x += a*w2.x; m2.y += a*w2.y; m2.z += a*w2.z; m2.w += a*w2.w;
        m3.x += a*w3.x; m3.y += a*w3.y; m3.z += a*w3.z; m3.w += a*w3.w;
    }
    float msg[16] = { m0.x,m0.y,m0.z,m0.w, m1.x,m1.y,m1.z,m1.w,
                      m2.x,m2.y,m2.z,m2.w, m3.x,m3.y,m3.z,m3.w };
    float* out = agg + (long)d * DD;
    #pragma unroll
    for (int j = 0; j < DD; ++j) atomicAdd(&out[j], msg[j]);
}

// ---------------- node update via WMMA: out = act(agg + h @ loop + b) ----------------
// One wave per 16-node tile; C/D 16x16 f32 = v8f per lane; K=16 as 4 x wmma_f32_16x16x4_f32.
// RELU is a compile-time flag so the epilogue lowers to v_max_num_f32 (no cndmask chains).
template <bool RELU>
__global__ void node_wmma_kernel(const float* __restrict__ h,     // [N,16]
                                 const float* __restrict__ loop,  // [16,16]
                                 const float* __restrict__ bias,  // [16]
                                 const float* __restrict__ agg,   // [N,16]
                                 float* __restrict__ out,         // [N,16]
                                 int ntiles) {
    int lane = threadIdx.x & 31;
    int tile = blockIdx.x * (blockDim.x >> 5) + (threadIdx.x >> 5);
    if (tile >= ntiles) return;                 // wave-uniform; EXEC stays all-ones
    int n0   = tile * 16;
    int col  = lane & 15;
    int half = lane >> 4;
    float bcol = bias[col];
    v8f c;
    #pragma unroll
    for (int i = 0; i < 8; ++i)                 // C init: agg tile + bias (row i+8*half, col)
        c[i] = agg[(long)(n0 + i + 8*half) * DD + col] + bcol;
    #pragma unroll
    for (int kc = 0; kc < 4; ++kc) {
        int k = kc * 4 + 2 * half;              // lanes 0-15: K=0,1 ; lanes 16-31: K=2,3
        v2f a, b;
        a.x = h[(long)(n0 + col) * DD + k];     // A: row = lane&15
        a.y = h[(long)(n0 + col) * DD + k + 1];
        b.x = loop[(k    ) * DD + col];         // B: N = lane&15, same K split
        b.y = loop[(k + 1) * DD + col];
        c = __builtin_amdgcn_wmma_f32_16x16x4_f32(false, a, false, b,
                                                  (short)0, c, false, false);
    }
    #pragma unroll
    for (int i = 0; i < 8; ++i) {
        float v = c[i];
        if (RELU) v = fmaxf(v, 0.0f);
        out[(long)(n0 + i + 8*half) * DD + col] = v;
    }
}

// ---------------- graph sum pooling ----------------
__global__ void pool_kernel(const float* __restrict__ h, const int* __restrict__ gid,
                            float* __restrict__ fp, int N) {
    int i = blockIdx.x * blockDim.x + threadIdx.x;
    if (i >= N * DD) return;
    int n = i >> 4, j = i & 15;
    atomicAdd(&fp[gid[n] * DD + j], h[i]);
}

// ---------------- attributor MLP layer 1: relu(fp[128,16] @ A0[16,64] + ab0) ----------------
__global__ void mlp1_kernel(const float* __restrict__ fp, const float* __restrict__ A0,
                            const float* __restrict__ ab0, float* __restrict__ fph) {
    int lane = threadIdx.x & 31;
    int col = lane & 15, half = lane >> 4;
    int mt = blockIdx.x >> 2, nt = blockIdx.x & 3;   // 8 x 4 tiles
    int m0 = mt * 16, nb = nt * 16;
    float bcol = ab0[nb + col];
    v8f c;
    #pragma unroll
    for (int i = 0; i < 8; ++i) c[i] = bcol;
    #pragma unroll
    for (int kc = 0; kc < 4; ++kc) {
        int k = kc * 4 + 2 * half;
        v2f a, b;
        a.x = fp[(m0 + col) * DD + k];
        a.y = fp[(m0 + col) * DD + k + 1];
        b.x = A0[(k    ) * ATTH + nb + col];
        b.y = A0[(k + 1) * ATTH + nb + col];
        c = __builtin_amdgcn_wmma_f32_16x16x4_f32(false, a, false, b,
                                                  (short)0, c, false, false);
    }
    #pragma unroll
    for (int i = 0; i < 8; ++i) {
        float v = fmaxf(c[i], 0.0f);
        fph[(m0 + i + 8*half) * ATTH + nb + col] = v;
    }
}

// ---------------- attributor MLP layer 2: sigmoid(fph[128,64] @ A1[64,32] + ab1) ----------------
__global__ void mlp2_kernel(const float* __restrict__ fph, const float* __restrict__ A1,
                            const float* __restrict__ ab1, float* __restrict__ out) {
    int lane = threadIdx.x & 31;
    int col = lane & 15, half = lane >> 4;
    int mt = blockIdx.x >> 1, nt = blockIdx.x & 1;   // 8 x 2 tiles
    int m0 = mt * 16, nb = nt * 16;
    float bcol = ab1[nb + col];
    v8f c;
    #pragma unroll
    for (int i = 0; i < 8; ++i) c[i] = bcol;
    #pragma unroll
    for (int kc = 0; kc < 16; ++kc) {                // K = 64
        int k = kc * 4 + 2 * half;
        v2f a, b;
        a.x = fph[(m0 + col) * ATTH + k];
        a.y = fph[(m0 + col) * ATTH + k + 1];
        b.x = A1[(k    ) * ATTO + nb + col];
        b.y = A1[(k + 1) * ATTO + nb + col];
        c = __builtin_amdgcn_wmma_f32_16x16x4_f32(false, a, false, b,
                                                  (short)0, c, false, false);
    }
    #pragma unroll
    for (int i = 0; i < 8; ++i) {
        float v = c[i];
        v = 1.0f / (1.0f + __expf(-v));
        out[(m0 + i + 8*half) * ATTO + nb + col] = v;
    }
}

extern "C" void kernel_launch(void* const* d_in, const int* in_sizes, int n_in,
                              void* d_out, int out_size, void* d_ws, size_t ws_size,
                              hipStream_t stream) {
    const int*   src   = (const int*)  d_in[0];
    const int*   dst   = (const int*)  d_in[1];
    const int*   et    = (const int*)  d_in[2];
    const int*   gid   = (const int*)  d_in[3];
    const float* W0    = (const float*)d_in[4];
    const float* loop0 = (const float*)d_in[5];
    const float* b0    = (const float*)d_in[6];
    const float* W1    = (const float*)d_in[7];
    const float* loop1 = (const float*)d_in[8];
    const float* b1    = (const float*)d_in[9];
    const float* W2    = (const float*)d_in[10];
    const float* loop2 = (const float*)d_in[11];
    const float* b2    = (const float*)d_in[12];
    const float* A0    = (const float*)d_in[13];
    const float* ab0   = (const float*)d_in[14];
    const float* A1    = (const float*)d_in[15];
    const float* ab1   = (const float*)d_in[16];
    int E = in_sizes[0];
    int N = in_sizes[3];
    int Np = (N + 15) & ~15;                    // 16-row padded

    // workspace layout (floats): deg[Np] | hA[16Np] | hB[16Np] | agg[16Np] | fp | fph
    float* ws  = (float*)d_ws;
    float* deg = ws;
    float* hA  = deg + Np;
    float* hB  = hA + (size_t)16 * Np;
    float* agg = hB + (size_t)16 * Np;
    float* fp  = agg + (size_t)16 * Np;
    float* fph = fp + NGRAPHS * DD;

    const int TB = 256;
    int eG   = (E + TB - 1) / TB;
    int nG16 = (N * DD + TB - 1) / TB;
    int ntiles = Np / 16;
    int nodeB  = 128;                            // 4 waves/block
    int nodeG  = (ntiles + (nodeB / 32) - 1) / (nodeB / 32);

    fill_zero<<<512,  TB, 0, stream>>>(deg, Np);
    fill_zero<<<2048, TB, 0, stream>>>(agg, (long)16 * Np);
    fill_zero<<<8,    TB, 0, stream>>>(fp, NGRAPHS * DD);

    deg_kernel    <<<eG,   TB, 0, stream>>>(dst, deg, E);
    l0_edge_kernel<<<eG,   TB, 0, stream>>>(deg, src, dst, et, W0, agg, E);
    l0_node_kernel<<<nG16, TB, 0, stream>>>(agg, deg, loop0, b0, hA, N);

    fill_zero<<<2048, TB, 0, stream>>>(agg, (long)16 * Np);
    rgcn_edge_kernel<<<eG, TB, 0, stream>>>(hA, src, dst, et, W1, agg, E);
    node_wmma_kernel<true><<<nodeG, nodeB, 0, stream>>>(hA, loop1, b1, agg, hB, ntiles);

    fill_zero<<<2048, TB, 0, stream>>>(agg, (long)16 * Np);
    rgcn_edge_kernel<<<eG, TB, 0, stream>>>(hB, src, dst, et, W2, agg, E);
    node_wmma_kernel<false><<<nodeG, nodeB, 0, stream>>>(hB, loop2, b2, agg, hA, ntiles);

    pool_kernel<<<nG16, TB, 0, stream>>>(hA, gid, fp, N);
    mlp1_kernel<<<32, 32, 0, stream>>>(fp, A0, ab0, fph);
    mlp2_kernel<<<16, 32, 0, stream>>>(fph, A1, ab1, (float*)d_out);
}